// GCN2_23691039605435
// MI455X (gfx1250) — compile-verified
//
#include <hip/hip_runtime.h>
#include <hip/hip_bf16.h>
#include <math.h>

typedef float v2f __attribute__((ext_vector_type(2)));
typedef float v8f __attribute__((ext_vector_type(8)));

#define THREADS 256

// ---------------- degree / symmetric norm ----------------
__global__ void k_deg_init(float* __restrict__ deg, int n) {
  int i = blockIdx.x * blockDim.x + threadIdx.x;
  if (i < n) deg[i] = 1.0f;                       // self-loop
}

__global__ void k_deg_count(const int* __restrict__ dst, float* __restrict__ deg, int E) {
  int e = blockIdx.x * blockDim.x + threadIdx.x;
  if (e < E) unsafeAtomicAdd(&deg[dst[e]], 1.0f); // global_atomic_add_f32
}

__global__ void k_deg_rsqrt(float* __restrict__ deg, int n) {
  int i = blockIdx.x * blockDim.x + threadIdx.x;
  if (i < n) deg[i] = rsqrtf(deg[i]);             // deg buffer becomes dinv
}

// ---------------- GEMM: C[nrows x 64] = act(A[nrows x 64]) @ W[64 x 64] ----------------
// One wave computes one 16x16 tile via 16x V_WMMA_F32_16X16X4_F32.
// Block = 8 waves = 2 row-tiles x 4 col-tiles = 32 rows x 64 cols.
template <bool RELU>
__global__ __launch_bounds__(THREADS) void k_gemm64(const float* __restrict__ Ain,
                                                    const float* __restrict__ W,
                                                    float* __restrict__ C, int nrows) {
  const int lane = threadIdx.x & 31;
  const int wave = threadIdx.x >> 5;
  const int rowTile = blockIdx.x * 2 + (wave >> 2);
  const int colTile = wave & 3;
  if (rowTile * 16 >= nrows) return;              // wave-uniform: EXEC stays all-1 for WMMA

  const int l16   = lane & 15;
  const int khalf = (lane >> 4) << 1;             // 0 for lanes 0-15, 2 for lanes 16-31
  int m = rowTile * 16 + l16;
  int mread = m < nrows ? m : nrows - 1;          // clamp tail reads, keep EXEC full
  const int n = colTile * 16 + l16;

  const float* __restrict__ arow = Ain + (size_t)mread * 64;
  v8f acc = {};

#pragma unroll
  for (int k0 = 0; k0 < 64; k0 += 4) {
    // A fragment (16x4 f32): lane holds K = k0+khalf, k0+khalf+1 of row m
    v2f a;
    a.x = arow[k0 + khalf];
    a.y = arow[k0 + khalf + 1];
    if (RELU) { a.x = fmaxf(a.x, 0.0f); a.y = fmaxf(a.y, 0.0f); }
    // B fragment (4x16 f32): lane holds col n of K-rows k0+khalf, k0+khalf+1
    v2f b;
    b.x = W[(size_t)(k0 + khalf) * 64 + n];
    b.y = W[(size_t)(k0 + khalf + 1) * 64 + n];
    acc = __builtin_amdgcn_wmma_f32_16x16x4_f32(false, a, false, b,
                                                (short)0, acc, false, false);
  }

  // C/D layout: VGPR v holds row M = v + 8*(lane>=16), col N = lane&15
  const int mo = (lane >> 4) * 8;
  float* __restrict__ cbase = C + (size_t)rowTile * 16 * 64 + (size_t)colTile * 16 + l16;
#pragma unroll
  for (int v = 0; v < 8; ++v) {
    int r = rowTile * 16 + v + mo;
    if (r < nrows) cbase[(size_t)(v + mo) * 64] = acc[v];
  }
}

// ---------------- aggregation ----------------
// A[i,d] = bias[d] + H[i,d] * dinv[i]^2   (self-loop message folded in)
__global__ void k_agg_init(const float* __restrict__ H, const float* __restrict__ dinv,
                           const float* __restrict__ bias, float* __restrict__ A, int n) {
  int idx = blockIdx.x * blockDim.x + threadIdx.x;
  if (idx < n * 64) {
    int i = idx >> 6, d = idx & 63;
    float di = dinv[i];
    A[idx] = bias[d] + H[idx] * di * di;
  }
}

// One wave per edge; each lane scatters a float2 of the 64-wide feature row.
__global__ __launch_bounds__(THREADS) void k_scatter(const float* __restrict__ H,
                                                     const int* __restrict__ src,
                                                     const int* __restrict__ dst,
                                                     const float* __restrict__ dinv,
                                                     float* __restrict__ A, int E) {
  int e = blockIdx.x * 8 + (threadIdx.x >> 5);
  if (e >= E) return;
  int lane = threadIdx.x & 31;
  int s = src[e], t = dst[e];
  float nrm = dinv[s] * dinv[t];
  const float2 h = *(const float2*)(H + (size_t)s * 64 + lane * 2);
  float* out = A + (size_t)t * 64 + lane * 2;
  unsafeAtomicAdd(out,     h.x * nrm);
  unsafeAtomicAdd(out + 1, h.y * nrm);
}

// ---------------- readout ----------------
__global__ void k_zero(float* acc) { acc[0] = 0.0f; }

// acc += sum_i sum_d relu(A[i,d]) * Wfc[d]; one wave per row
__global__ __launch_bounds__(THREADS) void k_reduce(const float* __restrict__ A,
                                                    const float* __restrict__ Wfc,
                                                    float* __restrict__ acc, int n) {
  __shared__ float ssum[8];
  int lane = threadIdx.x & 31;
  int wave = threadIdx.x >> 5;
  int row = blockIdx.x * 8 + wave;
  float p = 0.0f;
  if (row < n) {
    const float2 h = *(const float2*)(A + (size_t)row * 64 + lane * 2);
    const float2 w = *(const float2*)(Wfc + lane * 2);
    p = fmaxf(h.x, 0.0f) * w.x + fmaxf(h.y, 0.0f) * w.y;
  }
#pragma unroll
  for (int off = 16; off > 0; off >>= 1) p += __shfl_xor(p, off, 32);
  if (lane == 0) ssum[wave] = p;
  __syncthreads();
  if (threadIdx.x == 0) {
    float s = 0.0f;
#pragma unroll
    for (int w = 0; w < 8; ++w) s += ssum[w];
    unsafeAtomicAdd(acc, s);
  }
}

__global__ void k_final(const float* __restrict__ acc, const float* __restrict__ bfc,
                        float* __restrict__ out, float invN) {
  float v = acc[0] * invN + bfc[0];
  out[0] = 1.0f / (1.0f + expf(-v));
}

// ---------------- driver ----------------
extern "C" void kernel_launch(void* const* d_in, const int* in_sizes, int n_in,
                              void* d_out, int out_size, void* d_ws, size_t ws_size,
                              hipStream_t stream) {
  const float* x   = (const float*)d_in[0];
  const int*   ei  = (const int*)  d_in[1];
  const float* W1  = (const float*)d_in[2];
  const float* b1  = (const float*)d_in[3];
  const float* W2  = (const float*)d_in[4];
  const float* b2  = (const float*)d_in[5];
  const float* Wfc = (const float*)d_in[6];
  const float* bfc = (const float*)d_in[7];

  const int N = in_sizes[0] / 64;
  const int E = in_sizes[1] / 2;
  const int* src = ei;          // edge_index[0]
  const int* dst = ei + E;      // edge_index[1]

  // workspace layout (floats): dinv[Npad] | H[Npad*64] | A[Npad*64] | acc[1]
  float* ws = (float*)d_ws;
  const size_t Npad = (size_t)((N + 63) & ~63);
  float* dinv = ws;
  float* H    = ws + Npad;
  float* A    = H + Npad * 64;
  float* acc  = A + Npad * 64;

  const int rowTiles = (N + 15) / 16;
  const int gemmBlocks = (rowTiles + 1) / 2;
  const dim3 blk(THREADS);

  // degrees -> dinv
  k_deg_init <<<(N + THREADS - 1) / THREADS, blk, 0, stream>>>(dinv, N);
  k_deg_count<<<(E + THREADS - 1) / THREADS, blk, 0, stream>>>(dst, dinv, E);
  k_deg_rsqrt<<<(N + THREADS - 1) / THREADS, blk, 0, stream>>>(dinv, N);

  // layer 1: H = x @ W1 ; A = b1 + selfloop + scatter
  k_gemm64<false><<<gemmBlocks, blk, 0, stream>>>(x, W1, H, N);
  k_agg_init<<<((size_t)N * 64 + THREADS - 1) / THREADS, blk, 0, stream>>>(H, dinv, b1, A, N);
  k_scatter<<<(E + 7) / 8, blk, 0, stream>>>(H, src, dst, dinv, A, E);

  // layer 2: H = relu(A) @ W2 ; A = b2 + selfloop + scatter   (buffers reused)
  k_gemm64<true><<<gemmBlocks, blk, 0, stream>>>(A, W2, H, N);
  k_agg_init<<<((size_t)N * 64 + THREADS - 1) / THREADS, blk, 0, stream>>>(H, dinv, b2, A, N);
  k_scatter<<<(E + 7) / 8, blk, 0, stream>>>(H, src, dst, dinv, A, E);

  // readout: sigmoid(mean(relu(A) @ Wfc) + bfc)
  k_zero<<<1, 1, 0, stream>>>(acc);
  k_reduce<<<(N + 7) / 8, blk, 0, stream>>>(A, Wfc, acc, N);
  k_final<<<1, 1, 0, stream>>>(acc, bfc, (float*)d_out, 1.0f / (float)N);
}